// GATModel_29085518529138
// MI455X (gfx1250) — compile-verified
//
#include <hip/hip_runtime.h>
#include <hip/hip_bf16.h>

typedef __attribute__((ext_vector_type(2))) float v2f;
typedef __attribute__((ext_vector_type(8))) float v8f;

#define GAT_N   50000
#define GAT_E   800000
#define GAT_IN  256
#define GAT_HC  128   // H*C
#define GAT_H   4
#define TM      128
#define KT      32

// 16-byte int vector, as expected by the async-to-LDS builtin (per hipcc diag)
typedef int v4i __attribute__((vector_size(4 * sizeof(int))));
typedef __attribute__((address_space(1))) v4i gv4i;   // global
typedef __attribute__((address_space(3))) v4i lv4i;   // LDS

// ---------------- gfx1250 async global->LDS copy helpers ---------------------
__device__ __forceinline__ void async_copy_b128(const float* gsrc, float* lds_dst) {
#if __has_builtin(__builtin_amdgcn_global_load_async_to_lds_b128)
  __builtin_amdgcn_global_load_async_to_lds_b128(
      (gv4i*)gsrc, (lv4i*)lds_dst, 0, 0);
#else
  asm volatile("global_load_async_to_lds_b128 %0, %1, off"
               :: "v"((unsigned)(uintptr_t)lds_dst), "v"(gsrc)
               : "memory");
#endif
}

__device__ __forceinline__ void wait_async0() {
#if __has_builtin(__builtin_amdgcn_s_wait_asynccnt)
  __builtin_amdgcn_s_wait_asynccnt(0);
#else
  asm volatile("s_wait_asynccnt 0x0" ::: "memory");
#endif
}

// ---------------- monotone float <-> uint for atomicMax softmax-max ----------
__device__ __forceinline__ unsigned f2ord(float f) {
  unsigned u = __float_as_uint(f);
  return (u & 0x80000000u) ? ~u : (u | 0x80000000u);
}
__device__ __forceinline__ float ord2f(unsigned u) {
  return __uint_as_float((u & 0x80000000u) ? (u & 0x7FFFFFFFu) : ~u);
}

// ---------------- dense GEMM, f32 WMMA 16x16x4, out cols fixed at 128 --------
// A tile staged via async global->LDS (b128); W tile staged transposed so both
// A and B fragments are single ds_load_b64 per WMMA operand.
__global__ __launch_bounds__(256) void gemm_wmma_f32(
    const float* __restrict__ X, const float* __restrict__ W,
    float* __restrict__ XP, int M, int K)
{
  __shared__ float sA [TM][KT + 4];        // rows x k      (128 x 36)
  __shared__ float sBt[GAT_HC][KT + 4];    // cols x k      (128 x 36, transposed)

  const int t       = threadIdx.x;
  const int wave    = t >> 5;
  const int lane    = t & 31;
  const int half    = lane >> 4;      // 16-lane half selects the K pair
  const int l16     = lane & 15;
  const int mBase   = blockIdx.x * TM;
  const int rowBase = (wave >> 1) * 32;   // 0,32,64,96
  const int colBase = (wave & 1) * 64;    // 0,64

  v8f acc[2][4];
  for (int i = 0; i < 2; ++i)
    for (int j = 0; j < 4; ++j)
      acc[i][j] = (v8f){0.f, 0.f, 0.f, 0.f, 0.f, 0.f, 0.f, 0.f};

  for (int kc = 0; kc < K; kc += KT) {
    // ---- A tile (TM x KT): async b128 per lane, zeros for OOB rows ----
    for (int it = 0; it < 4; ++it) {
      int idx = it * 256 + t;             // 1024 float4 slots
      int r   = idx >> 3;                 // 8 float4 per row
      int c4  = (idx & 7) * 4;
      int gr  = mBase + r;
      if (gr < M) {
        async_copy_b128(X + (size_t)gr * K + kc + c4, &sA[r][c4]);
      } else {
        sA[r][c4 + 0] = 0.f; sA[r][c4 + 1] = 0.f;
        sA[r][c4 + 2] = 0.f; sA[r][c4 + 3] = 0.f;
      }
    }
    // ---- W tile (KT x 128): coalesced read, transposed store to LDS ----
    for (int it = 0; it < 4; ++it) {
      int idx = it * 256 + t;
      int kr  = idx >> 5;                 // 32 float4 per row
      int c4  = (idx & 31) * 4;
      float4 v = *(const float4*)(W + (size_t)(kc + kr) * GAT_HC + c4);
      sBt[c4 + 0][kr] = v.x; sBt[c4 + 1][kr] = v.y;
      sBt[c4 + 2][kr] = v.z; sBt[c4 + 3][kr] = v.w;
    }
    wait_async0();
    __syncthreads();

    for (int k = 0; k < KT; k += 4) {
      v2f a[2], b[4];
      // A 16x4: lanes 0-15 hold K=k,k+1 ; lanes 16-31 hold K=k+2,k+3
      for (int i = 0; i < 2; ++i) {
        const float* p = &sA[rowBase + i * 16 + l16][k + half * 2];
        a[i].x = p[0]; a[i].y = p[1];
      }
      // B 4x16: vgpr0 = rows {k,k+2} by half, vgpr1 = rows {k+1,k+3}
      for (int j = 0; j < 4; ++j) {
        const float* p = &sBt[colBase + j * 16 + l16][k + half * 2];
        b[j].x = p[0]; b[j].y = p[1];
      }
      for (int i = 0; i < 2; ++i)
        for (int j = 0; j < 4; ++j)
          acc[i][j] = __builtin_amdgcn_wmma_f32_16x16x4_f32(
              false, a[i], false, b[j], (short)0, acc[i][j], false, false);
    }
    __syncthreads();
  }

  // D layout: vgpr v, lanes 0-15 -> M=v ; lanes 16-31 -> M=v+8
  for (int i = 0; i < 2; ++i)
    for (int j = 0; j < 4; ++j) {
      int col = colBase + j * 16 + l16;
      for (int v = 0; v < 8; ++v) {
        int row = mBase + rowBase + i * 16 + v + half * 8;
        if (row < M) XP[(size_t)row * GAT_HC + col] = acc[i][j][v];
      }
    }
}

// ---------------- per-node attention scalars: alpha_s/alpha_d [N,H] ----------
__global__ __launch_bounds__(128) void alpha_kernel(
    const float* __restrict__ xp, const float* __restrict__ att_s,
    const float* __restrict__ att_d, float* __restrict__ aS, float* __restrict__ aD)
{
  int n = blockIdx.x;
  int t = threadIdx.x;              // t = h*32 + c ; one wave per head (wave32)
  float v  = xp[(size_t)n * GAT_HC + t];
  float ps = v * att_s[t];
  float pd = v * att_d[t];
  for (int off = 16; off > 0; off >>= 1) {
    ps += __shfl_xor(ps, off, 32);
    pd += __shfl_xor(pd, off, 32);
  }
  if ((t & 31) == 0) {
    int h = t >> 5;
    aS[(size_t)n * GAT_H + h] = ps;
    aD[(size_t)n * GAT_H + h] = pd;
  }
}

__device__ __forceinline__ void edge_src_dst(const int* ei, int e, int& src, int& dst) {
  if (e < GAT_E) { src = ei[e]; dst = ei[GAT_E + e]; }
  else           { src = dst = e - GAT_E; }           // self loop
}

__device__ __forceinline__ float edge_logit(const float* aS, const float* aD,
                                            int src, int dst, int h) {
  float s = aS[(size_t)src * GAT_H + h] + aD[(size_t)dst * GAT_H + h];
  return s > 0.f ? s : 0.2f * s;                      // leaky_relu(0.2)
}

// ---------------- edge pass 1: segment max over dst --------------------------
__global__ __launch_bounds__(256) void edge_max_kernel(
    const int* __restrict__ ei, const float* __restrict__ aS,
    const float* __restrict__ aD, unsigned* __restrict__ lmax, int workTotal)
{
  int gid = blockIdx.x * 256 + threadIdx.x;
  if (gid >= workTotal) return;
  int e = gid >> 2, h = gid & 3;
  int src, dst; edge_src_dst(ei, e, src, dst);
  float l = edge_logit(aS, aD, src, dst, h);
  atomicMax(lmax + (size_t)dst * GAT_H + h, f2ord(l));
}

// ---------------- edge pass 2: segment sum of exp ----------------------------
__global__ __launch_bounds__(256) void edge_sum_kernel(
    const int* __restrict__ ei, const float* __restrict__ aS,
    const float* __restrict__ aD, const unsigned* __restrict__ lmax,
    float* __restrict__ denom, int workTotal)
{
  int gid = blockIdx.x * 256 + threadIdx.x;
  if (gid >= workTotal) return;
  int e = gid >> 2, h = gid & 3;
  int src, dst; edge_src_dst(ei, e, src, dst);
  float l = edge_logit(aS, aD, src, dst, h);
  float ex = __expf(l - ord2f(lmax[(size_t)dst * GAT_H + h]));
  atomicAdd(denom + (size_t)dst * GAT_H + h, ex);
}

// ---------------- edge pass 3: weighted scatter of source features -----------
__global__ __launch_bounds__(128) void edge_agg_kernel(
    const int* __restrict__ ei, const float* __restrict__ aS,
    const float* __restrict__ aD, const unsigned* __restrict__ lmax,
    const float* __restrict__ denom, const float* __restrict__ xp,
    float* __restrict__ agg)
{
  int e = blockIdx.x;
  int t = threadIdx.x;              // t = h*32 + c
  int h = t >> 5;
  int src, dst; edge_src_dst(ei, e, src, dst);
  float l  = edge_logit(aS, aD, src, dst, h);
  float w  = __expf(l - ord2f(lmax[(size_t)dst * GAT_H + h])) /
             (denom[(size_t)dst * GAT_H + h] + 1e-16f);
  atomicAdd(agg + (size_t)dst * GAT_HC + t, xp[(size_t)src * GAT_HC + t] * w);
}

// ---------------- h = relu(agg + b) ------------------------------------------
__global__ __launch_bounds__(256) void bias_relu_kernel(
    const float* __restrict__ agg, const float* __restrict__ b,
    float* __restrict__ out, int total)
{
  int i = blockIdx.x * 256 + threadIdx.x;
  if (i >= total) return;
  float v = agg[i] + b[i & (GAT_HC - 1)];
  out[i] = v > 0.f ? v : 0.f;
}

// ---------------- out = (agg2 + b2)[root_index] ------------------------------
__global__ __launch_bounds__(256) void gather_kernel(
    const float* __restrict__ agg, const float* __restrict__ b,
    const int* __restrict__ root, float* __restrict__ out, int total)
{
  int i = blockIdx.x * 256 + threadIdx.x;
  if (i >= total) return;
  int r = i >> 7, t = i & (GAT_HC - 1);
  out[i] = agg[(size_t)root[r] * GAT_HC + t] + b[t];
}

extern "C" void kernel_launch(void* const* d_in, const int* in_sizes, int n_in,
                              void* d_out, int out_size, void* d_ws, size_t ws_size,
                              hipStream_t stream) {
  const float* x    = (const float*)d_in[0];
  const int*   ei   = (const int*)  d_in[1];
  const int*   root = (const int*)  d_in[2];
  const float* W1   = (const float*)d_in[3];
  const float* as1  = (const float*)d_in[4];
  const float* ad1  = (const float*)d_in[5];
  const float* b1   = (const float*)d_in[6];
  const float* W2   = (const float*)d_in[7];
  const float* as2  = (const float*)d_in[8];
  const float* ad2  = (const float*)d_in[9];
  const float* b2   = (const float*)d_in[10];

  const int Etot = GAT_E + GAT_N;                 // incl. self loops
  const size_t featBytes = (size_t)GAT_N * GAT_HC * sizeof(float);   // 25.6 MB
  const size_t nhBytes   = (size_t)GAT_N * GAT_H  * sizeof(float);   // 0.8 MB

  char* ws = (char*)d_ws;
  size_t off = 0;
  float*    xp    = (float*)(ws + off); off += featBytes;
  float*    agg   = (float*)(ws + off); off += featBytes;
  float*    hbuf  = (float*)(ws + off); off += featBytes;
  float*    aS    = (float*)(ws + off); off += nhBytes;
  float*    aD    = (float*)(ws + off); off += nhBytes;
  unsigned* lmax  = (unsigned*)(ws + off); off += nhBytes;
  float*    denom = (float*)(ws + off); off += nhBytes;
  // total ~80 MB of workspace used

  const int gemmGrid  = (GAT_N + TM - 1) / TM;
  const int workTotal = Etot * GAT_H;
  const int workGrid  = (workTotal + 255) / 256;
  const int featTotal = GAT_N * GAT_HC;
  const int featGrid  = (featTotal + 255) / 256;

  // ---------------- layer 1 ----------------
  gemm_wmma_f32<<<gemmGrid, 256, 0, stream>>>(x, W1, xp, GAT_N, GAT_IN);
  alpha_kernel<<<GAT_N, 128, 0, stream>>>(xp, as1, ad1, aS, aD);
  (void)hipMemsetAsync(lmax,  0, nhBytes,   stream);   // 0 == minimal ordered-float key
  (void)hipMemsetAsync(denom, 0, nhBytes,   stream);
  (void)hipMemsetAsync(agg,   0, featBytes, stream);
  edge_max_kernel<<<workGrid, 256, 0, stream>>>(ei, aS, aD, lmax, workTotal);
  edge_sum_kernel<<<workGrid, 256, 0, stream>>>(ei, aS, aD, lmax, denom, workTotal);
  edge_agg_kernel<<<Etot, 128, 0, stream>>>(ei, aS, aD, lmax, denom, xp, agg);
  bias_relu_kernel<<<featGrid, 256, 0, stream>>>(agg, b1, hbuf, featTotal);

  // ---------------- layer 2 ----------------
  gemm_wmma_f32<<<gemmGrid, 256, 0, stream>>>(hbuf, W2, xp, GAT_N, GAT_HC);
  alpha_kernel<<<GAT_N, 128, 0, stream>>>(xp, as2, ad2, aS, aD);
  (void)hipMemsetAsync(lmax,  0, nhBytes,   stream);
  (void)hipMemsetAsync(denom, 0, nhBytes,   stream);
  (void)hipMemsetAsync(agg,   0, featBytes, stream);
  edge_max_kernel<<<workGrid, 256, 0, stream>>>(ei, aS, aD, lmax, workTotal);
  edge_sum_kernel<<<workGrid, 256, 0, stream>>>(ei, aS, aD, lmax, denom, workTotal);
  edge_agg_kernel<<<Etot, 128, 0, stream>>>(ei, aS, aD, lmax, denom, xp, agg);

  // ---------------- root gather ----------------
  gather_kernel<<<(out_size + 255) / 256, 256, 0, stream>>>(agg, b2, root, (float*)d_out, out_size);
}